// BiDAF_71219147702356
// MI455X (gfx1250) — compile-verified
//
#include <hip/hip_runtime.h>
#include <math.h>

typedef __attribute__((ext_vector_type(2))) float v2f;
typedef __attribute__((ext_vector_type(8))) float v8f;

#define PAD_TOK 1
#define NEGV   -1e30f
#define Bsz 8
#define Tlen 1024
#define Jlen 64
#define Edim 100
#define Dh   128
#define D2   256
#define D3   384

__device__ __forceinline__ float sigm(float x) { return 1.f / (1.f + __expf(-x)); }

// ---------------------------------------------------------------------------
// Embedding gather with pad mask
// ---------------------------------------------------------------------------
__global__ void embed_kernel(const int* __restrict__ tok,
                             const float* __restrict__ table,
                             float* __restrict__ out, int NT) {
  long idx = (long)blockIdx.x * blockDim.x + threadIdx.x;
  long total = (long)NT * Edim;
  if (idx >= total) return;
  int p = (int)(idx / Edim);
  int e = (int)(idx - (long)p * Edim);
  int t = tok[p];
  out[idx] = (t == PAD_TOK) ? 0.f : table[(long)t * Edim + e];
}

// ---------------------------------------------------------------------------
// Tiled WMMA f32 GEMM:  C[z] = act( A[z](MxK) * B[z](NxK)^T + bias )
// M,N multiples of 64; K multiple of 4; all pointers 16B aligned.
// Block 256 threads, 64x64 tile, K chunked by 32.
// ---------------------------------------------------------------------------
__global__ void gemm_nt_kernel(const float* __restrict__ A, long sA,
                               const float* __restrict__ Bm, long sB,
                               float* __restrict__ C, long sC,
                               int M, int N, int K,
                               const float* __restrict__ bias, int act) {
  __shared__ __align__(16) float As[64][36];   // 36 floats = 144B (16B multiple)
  __shared__ __align__(16) float Bs[64][36];
  const int z = blockIdx.z;
  A  += (long)z * sA;
  Bm += (long)z * sB;
  C  += (long)z * sC;
  const int tM = blockIdx.y * 64, tN = blockIdx.x * 64;
  const int tid = threadIdx.x;
  const int lane = tid & 31, wave = tid >> 5;
  const int wm = wave & 3;       // M sub-tile
  const int wn = wave >> 2;      // N half (0/1) -> offset wn*32
  const int l = lane & 15, h = lane >> 4;
  const int lr = tid >> 2;            // 0..63
  const int lc = (tid & 3) * 8;       // 0,8,16,24
  const float* Arow = A + (long)(tM + lr) * K;
  const float* Brow = Bm + (long)(tN + lr) * K;
  v8f acc0 = {}; v8f acc1 = {};
  for (int k0 = 0; k0 < K; k0 += 32) {
    if (k0 + 32 <= K) {
      float4 a0 = *(const float4*)(Arow + k0 + lc);
      float4 a1 = *(const float4*)(Arow + k0 + lc + 4);
      float4 b0 = *(const float4*)(Brow + k0 + lc);
      float4 b1 = *(const float4*)(Brow + k0 + lc + 4);
      *(float4*)&As[lr][lc]     = a0;
      *(float4*)&As[lr][lc + 4] = a1;
      *(float4*)&Bs[lr][lc]     = b0;
      *(float4*)&Bs[lr][lc + 4] = b1;
      if (k0 + 64 <= K) {
        __builtin_prefetch(Arow + k0 + 32 + lc, 0, 1);
        __builtin_prefetch(Brow + k0 + 32 + lc, 0, 1);
      }
    } else {
#pragma unroll
      for (int j = 0; j < 8; ++j) {
        int kk = k0 + lc + j;
        float av = 0.f, bv = 0.f;
        if (kk < K) { av = Arow[kk]; bv = Brow[kk]; }
        As[lr][lc + j] = av;
        Bs[lr][lc + j] = bv;
      }
    }
    __syncthreads();
#pragma unroll
    for (int kk = 0; kk < 32; kk += 4) {
      v2f a, b0, b1;
      a.x  = As[wm * 16 + l][kk + 2 * h];
      a.y  = As[wm * 16 + l][kk + 2 * h + 1];
      b0.x = Bs[wn * 32 + l][kk + 2 * h];
      b0.y = Bs[wn * 32 + l][kk + 2 * h + 1];
      b1.x = Bs[wn * 32 + 16 + l][kk + 2 * h];
      b1.y = Bs[wn * 32 + 16 + l][kk + 2 * h + 1];
      acc0 = __builtin_amdgcn_wmma_f32_16x16x4_f32(false, a, false, b0, (short)0, acc0, false, false);
      acc1 = __builtin_amdgcn_wmma_f32_16x16x4_f32(false, a, false, b1, (short)0, acc1, false, false);
    }
    __syncthreads();
  }
#pragma unroll
  for (int r = 0; r < 8; ++r) {
    int gm = tM + wm * 16 + r + 8 * h;
    int n0 = tN + wn * 32 + l;
    int n1 = n0 + 16;
    float v0 = acc0[r], v1 = acc1[r];
    if (bias) { v0 += bias[n0]; v1 += bias[n1]; }
    if (act == 1) { v0 = tanhf(v0); v1 = tanhf(v1); }
    C[(long)gm * N + n0] = v0;
    C[(long)gm * N + n1] = v1;
  }
}

// ---------------------------------------------------------------------------
// Tiled WMMA f32 GEMM:  C[z] = A[z](MxK) * B[z](KxN)
// ---------------------------------------------------------------------------
__global__ void gemm_nn_kernel(const float* __restrict__ A, long sA,
                               const float* __restrict__ Bm, long sB,
                               float* __restrict__ C, long sC,
                               int M, int N, int K) {
  __shared__ __align__(16) float As[64][36];
  __shared__ __align__(16) float Bs[32][68];   // 68 floats = 272B (16B multiple)
  const int z = blockIdx.z;
  A  += (long)z * sA;
  Bm += (long)z * sB;
  C  += (long)z * sC;
  const int tM = blockIdx.y * 64, tN = blockIdx.x * 64;
  const int tid = threadIdx.x;
  const int lane = tid & 31, wave = tid >> 5;
  const int wm = wave & 3;
  const int wn = wave >> 2;
  const int l = lane & 15, h = lane >> 4;
  const int lr = tid >> 2;        // 0..63
  const int lc = (tid & 3) * 8;   // A cols
  const int kr = tid >> 3;        // 0..31
  const int kc = (tid & 7) * 8;   // B cols
  const float* Arow = A + (long)(tM + lr) * K;
  v8f acc0 = {}; v8f acc1 = {};
  for (int k0 = 0; k0 < K; k0 += 32) {
    const float* Brow = Bm + (long)(k0 + kr) * N + tN + kc;
    if (k0 + 32 <= K) {
      float4 a0 = *(const float4*)(Arow + k0 + lc);
      float4 a1 = *(const float4*)(Arow + k0 + lc + 4);
      float4 b0 = *(const float4*)(Brow);
      float4 b1 = *(const float4*)(Brow + 4);
      *(float4*)&As[lr][lc]     = a0;
      *(float4*)&As[lr][lc + 4] = a1;
      *(float4*)&Bs[kr][kc]     = b0;
      *(float4*)&Bs[kr][kc + 4] = b1;
      if (k0 + 64 <= K) {
        __builtin_prefetch(Arow + k0 + 32 + lc, 0, 1);
        __builtin_prefetch(Brow + (long)32 * N, 0, 1);
      }
    } else {
#pragma unroll
      for (int j = 0; j < 8; ++j) {
        int kk = k0 + lc + j;
        As[lr][lc + j] = (kk < K) ? Arow[kk] : 0.f;
      }
#pragma unroll
      for (int j = 0; j < 8; ++j)
        Bs[kr][kc + j] = (k0 + kr < K) ? Brow[j] : 0.f;
    }
    __syncthreads();
#pragma unroll
    for (int kk = 0; kk < 32; kk += 4) {
      v2f a, b0, b1;
      a.x  = As[wm * 16 + l][kk + 2 * h];
      a.y  = As[wm * 16 + l][kk + 2 * h + 1];
      b0.x = Bs[kk + 2 * h][wn * 32 + l];
      b0.y = Bs[kk + 2 * h + 1][wn * 32 + l];
      b1.x = Bs[kk + 2 * h][wn * 32 + 16 + l];
      b1.y = Bs[kk + 2 * h + 1][wn * 32 + 16 + l];
      acc0 = __builtin_amdgcn_wmma_f32_16x16x4_f32(false, a, false, b0, (short)0, acc0, false, false);
      acc1 = __builtin_amdgcn_wmma_f32_16x16x4_f32(false, a, false, b1, (short)0, acc1, false, false);
    }
    __syncthreads();
  }
#pragma unroll
  for (int r = 0; r < 8; ++r) {
    int gm = tM + wm * 16 + r + 8 * h;
    int n0 = tN + wn * 32 + l;
    C[(long)gm * N + n0] = acc0[r];
    C[(long)gm * N + n0 + 16] = acc1[r];
  }
}

// ---------------------------------------------------------------------------
// y[m] = act( A1[m,:K1].w1 + A2[m,:K2].w2 + b ); one wave per row, float4.
// K1,K2 multiples of 4; rows 16B aligned.
// ---------------------------------------------------------------------------
__global__ void rowdot_kernel(const float* __restrict__ A1, int K1, const float* __restrict__ w1,
                              const float* __restrict__ A2, int K2, const float* __restrict__ w2,
                              const float* __restrict__ bias, int act,
                              float* __restrict__ y, int Mtot) {
  int row = blockIdx.x * (blockDim.x >> 5) + (threadIdx.x >> 5);
  int lane = threadIdx.x & 31;
  if (row >= Mtot) return;
  float s = 0.f;
  const float* a = A1 + (long)row * K1;
  for (int k = lane * 4; k < K1; k += 128) {
    float4 av = *(const float4*)(a + k);
    float4 wv = *(const float4*)(w1 + k);
    s += av.x * wv.x + av.y * wv.y + av.z * wv.z + av.w * wv.w;
  }
  if (A2) {
    const float* a2 = A2 + (long)row * K2;
    for (int k = lane * 4; k < K2; k += 128) {
      float4 av = *(const float4*)(a2 + k);
      float4 wv = *(const float4*)(w2 + k);
      s += av.x * wv.x + av.y * wv.y + av.z * wv.z + av.w * wv.w;
    }
  }
  for (int off = 16; off; off >>= 1) s += __shfl_xor(s, off, 32);
  if (lane == 0) {
    if (bias) s += bias[0];
    if (act == 1) s = sigm(s);
    y[row] = s;
  }
}

// ---------------------------------------------------------------------------
// Bidirectional GRU scan. Block 0 = forward, block 1 = backward (time-rev).
// d=128, B=8 fixed; out row stride outLD, offset dir*128.
// ---------------------------------------------------------------------------
__global__ void gru_scan_kernel(const float* __restrict__ xgF, const float* __restrict__ xgB,
                                const float* __restrict__ WhhF, const float* __restrict__ bhhF,
                                const float* __restrict__ WhhB, const float* __restrict__ bhhB,
                                float* __restrict__ out, int T, int outLD) {
  const int dir = blockIdx.x;
  const float* xg  = dir ? xgB  : xgF;
  const float* Whh = dir ? WhhB : WhhF;
  const float* bhh = dir ? bhhB : bhhF;
  const int outOff = dir * Dh;
  __shared__ __align__(16) float hsh[Bsz * 132];   // 132 floats = 528B (16B mult)
  __shared__ float ghs[Bsz * 392];
  const int tid = threadIdx.x;   // 512
  for (int o = tid; o < Bsz * 132; o += 512) hsh[o] = 0.f;
  __syncthreads();
  for (int s = 0; s < T; ++s) {
    int t = dir ? (T - 1 - s) : s;
#pragma unroll
    for (int u = 0; u < 6; ++u) {
      int o = tid + u * 512;           // 0..3071
      int row = o >> 3, b = o & 7;
      const float4* wr4 = (const float4*)(Whh + (long)row * Dh);
      const float4* hb4 = (const float4*)(hsh + b * 132);
      float sum = 0.f;
#pragma unroll 8
      for (int k = 0; k < Dh / 4; ++k) {
        float4 wv = wr4[k];
        float4 hv = hb4[k];
        sum += wv.x * hv.x + wv.y * hv.y + wv.z * hv.z + wv.w * hv.w;
      }
      ghs[b * 392 + row] = sum + bhh[row];
    }
    __syncthreads();
#pragma unroll
    for (int u = 0; u < 2; ++u) {
      int o = tid + u * 512;           // 0..1023
      int b = o >> 7, i = o & 127;
      const float* xr = xg + (long)(b * T + t) * D3;
      float hr = ghs[b * 392 + i];
      float hz = ghs[b * 392 + Dh + i];
      float hn = ghs[b * 392 + 2 * Dh + i];
      float r = sigm(xr[i] + hr);
      float z = sigm(xr[Dh + i] + hz);
      float n = tanhf(xr[2 * Dh + i] + r * hn);
      float hold = hsh[b * 132 + i];
      float hnew = (1.f - z) * n + z * hold;
      hsh[b * 132 + i] = hnew;
      out[(long)(b * T + t) * outLD + outOff + i] = hnew;
    }
    __syncthreads();
  }
}

// ---------------------------------------------------------------------------
// Elementwise helpers
// ---------------------------------------------------------------------------
__global__ void scalevec_kernel(const float* __restrict__ x, const float* __restrict__ w,
                                float* __restrict__ out, long n) {
  long idx = (long)blockIdx.x * blockDim.x + threadIdx.x;
  if (idx >= n) return;
  out[idx] = x[idx] * w[idx & (D2 - 1)];
}

__global__ void s_assemble_kernel(float* __restrict__ S, const float* __restrict__ sc,
                                  const float* __restrict__ sq, const int* __restrict__ question) {
  long idx = (long)blockIdx.x * blockDim.x + threadIdx.x;
  if (idx >= (long)Bsz * Tlen * Jlen) return;
  int j = (int)(idx & (Jlen - 1));
  long bt = idx >> 6;
  int b = (int)(bt >> 10);
  bool np = question[b * Jlen + j] != PAD_TOK;
  S[idx] = np ? (S[idx] + sc[bt] + sq[b * Jlen + j]) : NEGV;
}

// softmax over J=64 per row (one wave / row) + masked row max -> maxtmp
__global__ void softmaxJ_kernel(float* __restrict__ S, float* __restrict__ maxtmp,
                                const int* __restrict__ article) {
  int row = blockIdx.x * (blockDim.x >> 5) + (threadIdx.x >> 5);
  if (row >= Bsz * Tlen) return;
  int lane = threadIdx.x & 31;
  float* r = S + (long)row * Jlen;
  float v0 = r[lane], v1 = r[lane + 32];
  float m = fmaxf(v0, v1);
  for (int off = 16; off; off >>= 1) m = fmaxf(m, __shfl_xor(m, off, 32));
  if (lane == 0) {
    int b = row >> 10, t = row & (Tlen - 1);
    maxtmp[row] = (article[b * Tlen + t] != PAD_TOK) ? m : NEGV;
  }
  float e0 = __expf(v0 - m), e1 = __expf(v1 - m);
  float s = e0 + e1;
  for (int off = 16; off; off >>= 1) s += __shfl_xor(s, off, 32);
  float inv = 1.f / s;
  r[lane] = e0 * inv;
  r[lane + 32] = e1 * inv;
}

// softmax over a 1024-length row; optional column mask by article pad.
__global__ void softmax1024_kernel(float* __restrict__ X, int rowsPerBatch,
                                   const int* __restrict__ tok) {
  __shared__ float red[8];
  int row = blockIdx.x;
  int b = row / rowsPerBatch;
  float* x = X + (long)row * 1024;
  int tid = threadIdx.x;      // 256
  float vals[4];
  float m = -INFINITY;
#pragma unroll
  for (int i = 0; i < 4; ++i) {
    int j = tid + i * 256;
    float v = x[j];
    if (tok && tok[(long)b * 1024 + j] == PAD_TOK) v = NEGV;
    vals[i] = v;
    m = fmaxf(m, v);
  }
  for (int off = 16; off; off >>= 1) m = fmaxf(m, __shfl_xor(m, off, 32));
  if ((tid & 31) == 0) red[tid >> 5] = m;
  __syncthreads();
  float m2 = red[0];
#pragma unroll
  for (int w = 1; w < 8; ++w) m2 = fmaxf(m2, red[w]);
  __syncthreads();
  float s = 0.f;
#pragma unroll
  for (int i = 0; i < 4; ++i) { vals[i] = __expf(vals[i] - m2); s += vals[i]; }
  for (int off = 16; off; off >>= 1) s += __shfl_xor(s, off, 32);
  if ((tid & 31) == 0) red[tid >> 5] = s;
  __syncthreads();
  float s2 = 0.f;
#pragma unroll
  for (int w = 0; w < 8; ++w) s2 += red[w];
  float inv = 1.f / s2;
#pragma unroll
  for (int i = 0; i < 4; ++i) x[tid + i * 256] = vals[i] * inv;
}

__global__ void q2c_kernel(const float* __restrict__ batt, const float* __restrict__ ec,
                           float* __restrict__ q2c) {
  int b = blockIdx.x;
  int i = threadIdx.x;  // 256
  float s = 0.f;
  for (int t = 0; t < Tlen; ++t)
    s += batt[b * Tlen + t] * ec[(long)(b * Tlen + t) * D2 + i];
  q2c[b * D2 + i] = s;
}

__global__ void g_assemble_kernel(const float* __restrict__ ec, const float* __restrict__ c2q,
                                  const float* __restrict__ q2c, float* __restrict__ G) {
  long idx = (long)blockIdx.x * blockDim.x + threadIdx.x;
  if (idx >= (long)Bsz * Tlen * D2) return;
  long bt = idx >> 8;
  int i = (int)(idx & (D2 - 1));
  int b = (int)(bt >> 10);
  float e = ec[idx], c = c2q[idx], q = q2c[b * D2 + i];
  float* g = G + bt * (4 * D2);
  g[i] = e; g[D2 + i] = c; g[2 * D2 + i] = e * c; g[3 * D2 + i] = e * q;
}

__global__ void x_assemble_kernel(const float* __restrict__ M, const float* __restrict__ Mh,
                                  float* __restrict__ X) {
  long idx = (long)blockIdx.x * blockDim.x + threadIdx.x;
  if (idx >= (long)Bsz * Tlen * D2) return;
  long bt = idx >> 8;
  int i = (int)(idx & (D2 - 1));
  float a = M[idx], h = Mh[idx];
  float* xr = X + bt * (4 * D2);
  xr[i] = a; xr[D2 + i] = h; xr[2 * D2 + i] = a * h; xr[3 * D2 + i] = a - h;
}

__global__ void fuse_kernel(const float* __restrict__ g, const float* __restrict__ m,
                            const float* __restrict__ M, float* __restrict__ out) {
  long idx = (long)blockIdx.x * blockDim.x + threadIdx.x;
  if (idx >= (long)Bsz * Tlen * D2) return;
  long bt = idx >> 8;
  float gg = g[bt];
  out[idx] = gg * m[idx] + (1.f - gg) * M[idx];
}

// ---------------------------------------------------------------------------
// Host orchestration
// ---------------------------------------------------------------------------
static void gru_layer_ptrs_packed(const float* base, const int* ins, int nlayers,
                                  const float** out) {
  size_t off = 0;
  for (int l = 0; l < nlayers; ++l)
    for (int dir = 0; dir < 2; ++dir) {
      out[l * 8 + dir * 4 + 0] = base + off; off += (size_t)D3 * ins[l];
      out[l * 8 + dir * 4 + 1] = base + off; off += (size_t)D3 * Dh;
      out[l * 8 + dir * 4 + 2] = base + off; off += D3;
      out[l * 8 + dir * 4 + 3] = base + off; off += D3;
    }
}

extern "C" void kernel_launch(void* const* d_in, const int* in_sizes, int n_in,
                              void* d_out, int out_size, void* d_ws, size_t ws_size,
                              hipStream_t stream) {
  (void)in_sizes; (void)out_size; (void)ws_size;
  const bool flat = (n_in >= 60);   // each list element its own pointer (n_in==69)

  const int*   article = (const int*)d_in[0];
  const int*   question = (const int*)d_in[1];
  const float* embed_table = (const float*)d_in[2];

  const float* ctx[8];  const float* model[16];
  const float* comp[16]; const float* p2g[16];
  const float *w_sim, *self_W, *fuse_W, *fuse_b, *gate_W, *gate_b;
  const float *p1_W, *p1_b, *p2_W, *p2_b;

  if (flat) {
    for (int i = 0; i < 8;  ++i) ctx[i]   = (const float*)d_in[3 + i];
    w_sim = (const float*)d_in[11];
    for (int i = 0; i < 16; ++i) model[i] = (const float*)d_in[12 + i];
    self_W = (const float*)d_in[28];
    fuse_W = (const float*)d_in[29]; fuse_b = (const float*)d_in[30];
    gate_W = (const float*)d_in[31]; gate_b = (const float*)d_in[32];
    for (int i = 0; i < 16; ++i) comp[i]  = (const float*)d_in[33 + i];
    p1_W = (const float*)d_in[49]; p1_b = (const float*)d_in[50];
    for (int i = 0; i < 16; ++i) p2g[i]   = (const float*)d_in[51 + i];
    p2_W = (const float*)d_in[67]; p2_b = (const float*)d_in[68];
  } else {
    int insCtx[1]  = {Edim};
    int insModel[2] = {4 * D2, D2};
    int ins22[2]   = {D2, D2};
    gru_layer_ptrs_packed((const float*)d_in[3], insCtx, 1, ctx);
    w_sim = (const float*)d_in[4];
    gru_layer_ptrs_packed((const float*)d_in[5], insModel, 2, model);
    self_W = (const float*)d_in[6];
    fuse_W = (const float*)d_in[7]; fuse_b = (const float*)d_in[8];
    gate_W = (const float*)d_in[9]; gate_b = (const float*)d_in[10];
    gru_layer_ptrs_packed((const float*)d_in[11], ins22, 2, comp);
    p1_W = (const float*)d_in[12]; p1_b = (const float*)d_in[13];
    gru_layer_ptrs_packed((const float*)d_in[14], ins22, 2, p2g);
    p2_W = (const float*)d_in[15]; p2_b = (const float*)d_in[16];
  }

  // ---- workspace arena (floats) ----
  float* w = (float*)d_ws;
  size_t off = 0;
  auto alloc = [&](size_t n) { float* p = w + off; off += n; return p; };
  const size_t BT = (size_t)Bsz * Tlen;     // 8192
  const size_t BJ = (size_t)Bsz * Jlen;     // 512
  float* embA  = alloc(BT * Edim);
  float* embQ  = alloc(BJ * Edim);
  float* xgF   = alloc(BT * D3);
  float* xgB   = alloc(BT * D3);
  float* xgFq  = alloc(BJ * D3);
  float* xgBq  = alloc(BJ * D3);
  float* ec    = alloc(BT * D2);
  float* eq    = alloc(BJ * D2);
  float* scv   = alloc(BT);
  float* sqv   = alloc(BJ);
  float* ecw   = alloc(BT * D2);
  float* S     = alloc(BT * Jlen);
  float* maxtmp= alloc(BT);
  float* batt  = alloc(BT);
  float* q2cB  = alloc((size_t)Bsz * D2);
  float* c2q   = alloc(BT * D2);
  float* G     = alloc(BT * 4 * D2);
  float* tmpA  = alloc(BT * D2);
  float* Mm    = alloc(BT * D2);
  float* MW    = alloc(BT * D2);
  float* score = alloc(BT * Tlen);
  float* Mhat  = alloc(BT * D2);
  float* X     = alloc(BT * 4 * D2);
  float* mb    = alloc(BT * D2);
  float* gb    = alloc(BT);
  float* Mfuse = alloc(BT * D2);
  float* Mc    = alloc(BT * D2);
  float* M2    = alloc(BT * D2);

  auto gemmNT = [&](const float* A, long sA, const float* Bp, long sB, float* C, long sC,
                    int M, int N, int K, const float* bias, int act, int zb) {
    dim3 grid(N / 64, M / 64, zb);
    gemm_nt_kernel<<<grid, 256, 0, stream>>>(A, sA, Bp, sB, C, sC, M, N, K, bias, act);
  };
  auto gemmNN = [&](const float* A, long sA, const float* Bp, long sB, float* C, long sC,
                    int M, int N, int K, int zb) {
    dim3 grid(N / 64, M / 64, zb);
    gemm_nn_kernel<<<grid, 256, 0, stream>>>(A, sA, Bp, sB, C, sC, M, N, K);
  };
  auto biGru = [&](const float* x, int Mrows, int Tl, int K, const float* const* p, float* out) {
    float* xf = (Tl == Tlen) ? xgF : xgFq;
    float* xb = (Tl == Tlen) ? xgB : xgBq;
    gemmNT(x, 0, p[0], 0, xf, 0, Mrows, D3, K, p[2], 0, 1);
    gemmNT(x, 0, p[4], 0, xb, 0, Mrows, D3, K, p[6], 0, 1);
    gru_scan_kernel<<<2, 512, 0, stream>>>(xf, xb, p[1], p[3], p[5], p[7], out, Tl, D2);
  };

  // 1) embeddings
  embed_kernel<<<(unsigned)((BT * Edim + 255) / 256), 256, 0, stream>>>(article, embed_table, embA, (int)BT);
  embed_kernel<<<(unsigned)((BJ * Edim + 255) / 256), 256, 0, stream>>>(question, embed_table, embQ, (int)BJ);

  // 2) context GRU
  biGru(embA, (int)BT, Tlen, Edim, ctx, ec);
  biGru(embQ, (int)BJ, Jlen, Edim, ctx, eq);

  // 3) similarity + attention
  rowdot_kernel<<<(int)((BT + 7) / 8), 256, 0, stream>>>(ec, D2, w_sim, nullptr, 0, nullptr, nullptr, 0, scv, (int)BT);
  rowdot_kernel<<<(int)((BJ + 7) / 8), 256, 0, stream>>>(eq, D2, w_sim + D2, nullptr, 0, nullptr, nullptr, 0, sqv, (int)BJ);
  scalevec_kernel<<<(unsigned)((BT * D2 + 255) / 256), 256, 0, stream>>>(ec, w_sim + 2 * D2, ecw, (long)BT * D2);
  gemmNT(ecw, (long)Tlen * D2, eq, (long)Jlen * D2, S, (long)Tlen * Jlen, Tlen, Jlen, D2, nullptr, 0, Bsz);
  s_assemble_kernel<<<(unsigned)((BT * Jlen + 255) / 256), 256, 0, stream>>>(S, scv, sqv, question);
  softmaxJ_kernel<<<(int)((BT + 7) / 8), 256, 0, stream>>>(S, maxtmp, article);
  softmax1024_kernel<<<Bsz, 256, 0, stream>>>(maxtmp, 1, nullptr);           // -> batt in place
  (void)batt;
  gemmNN(S, (long)Tlen * Jlen, eq, (long)Jlen * D2, c2q, (long)Tlen * D2, Tlen, D2, Jlen, Bsz);
  q2c_kernel<<<Bsz, D2, 0, stream>>>(maxtmp, ec, q2cB);
  g_assemble_kernel<<<(unsigned)((BT * D2 + 255) / 256), 256, 0, stream>>>(ec, c2q, q2cB, G);

  // 4) model GRU (2 layers)
  biGru(G, (int)BT, Tlen, 4 * D2, model, tmpA);
  biGru(tmpA, (int)BT, Tlen, D2, model + 8, Mm);

  // 5) self attention
  gemmNT(Mm, 0, self_W, 0, MW, 0, (int)BT, D2, D2, nullptr, 0, 1);
  gemmNT(MW, (long)Tlen * D2, Mm, (long)Tlen * D2, score, (long)Tlen * Tlen, Tlen, Tlen, D2, nullptr, 0, Bsz);
  softmax1024_kernel<<<(unsigned)BT, 256, 0, stream>>>(score, Tlen, article);
  gemmNN(score, (long)Tlen * Tlen, Mm, (long)Tlen * D2, Mhat, (long)Tlen * D2, Tlen, D2, Tlen, Bsz);

  // 6) gated fusion
  x_assemble_kernel<<<(unsigned)((BT * D2 + 255) / 256), 256, 0, stream>>>(Mm, Mhat, X);
  gemmNT(X, 0, fuse_W, 0, mb, 0, (int)BT, D2, 4 * D2, fuse_b, 1, 1);
  rowdot_kernel<<<(int)((BT + 7) / 8), 256, 0, stream>>>(X, 4 * D2, gate_W, nullptr, 0, nullptr, gate_b, 1, gb, (int)BT);
  fuse_kernel<<<(unsigned)((BT * D2 + 255) / 256), 256, 0, stream>>>(gb, mb, Mm, Mfuse);

  // 7) comp GRU + p1
  biGru(Mfuse, (int)BT, Tlen, D2, comp, tmpA);
  biGru(tmpA, (int)BT, Tlen, D2, comp + 8, Mc);
  float* out = (float*)d_out;
  rowdot_kernel<<<(int)((BT + 7) / 8), 256, 0, stream>>>(G, 4 * D2, p1_W, Mc, D2, p1_W + 4 * D2, p1_b, 0, out, (int)BT);

  // 8) p2 GRU + p2
  biGru(Mc, (int)BT, Tlen, D2, p2g, tmpA);
  biGru(tmpA, (int)BT, Tlen, D2, p2g + 8, M2);
  rowdot_kernel<<<(int)((BT + 7) / 8), 256, 0, stream>>>(G, 4 * D2, p2_W, M2, D2, p2_W + 4 * D2, p2_b, 0, out + BT, (int)BT);
}